// ProjectionBasedGate_8735963480504
// MI455X (gfx1250) — compile-verified
//
#include <hip/hip_runtime.h>
#include <math.h>

typedef __attribute__((ext_vector_type(2))) float v2f;
typedef __attribute__((ext_vector_type(8))) float v8f;

// Problem constants (fixed by the reference).
#define NTOK 8192   // 4*2048 tokens
#define DDIM 1024
#define NEXP 8
#define KDIM 64
#define NC   512    // NEXP*KDIM

// ---------------------------------------------------------------------------
// Kernel 1: C[n, e*64+j] = sum_d X[n,d] * V[e,d,j]   (the big 8.6 GFLOP GEMM)
// Block tile 64(M) x 32(N), K-chunk 32. 8 waves, one 16x16 WMMA tile each.
// ---------------------------------------------------------------------------
__global__ __launch_bounds__(256) void k_gemm_xv(const float* __restrict__ X,
                                                 const float* __restrict__ V,
                                                 float* __restrict__ C) {
  __shared__ float As[64][36];   // pad 36: 16B-aligned rows, conflict-free frag reads
  __shared__ float Bs[32][32];
  const int m0 = blockIdx.x * 64;
  const int n0 = blockIdx.y * 32;
  const int e  = n0 >> 6;        // expert owning this N-tile
  const int jb = n0 & 63;        // column base within expert
  const int tid = threadIdx.x;
  const int w = tid >> 5, lane = tid & 31;
  const int wm = w & 3, wn = w >> 2;
  const int half = lane >> 4, lr = lane & 15;
  const int arow = tid >> 2, acb = (tid & 3) * 8;
  const int brow = tid >> 3, bcb = (tid & 7) * 4;
  v8f acc = {};
  for (int kt = 0; kt < DDIM; kt += 32) {
    const float4* asrc = (const float4*)&X[(size_t)(m0 + arow) * DDIM + kt + acb];
    float4 av0 = asrc[0], av1 = asrc[1];
    float4 bv  = *(const float4*)&V[((size_t)e * DDIM + kt + brow) * KDIM + jb + bcb];
    *(float4*)&As[arow][acb]     = av0;
    *(float4*)&As[arow][acb + 4] = av1;
    *(float4*)&Bs[brow][bcb]     = bv;
    __syncthreads();
#pragma unroll
    for (int kk = 0; kk < 32; kk += 4) {
      const int krow = kk + 2 * half;       // lanes 0-15: K=0,1 ; lanes 16-31: K=2,3
      v2f a, b;
      a[0] = As[wm * 16 + lr][krow];
      a[1] = As[wm * 16 + lr][krow + 1];
      b[0] = Bs[krow][wn * 16 + lr];
      b[1] = Bs[krow + 1][wn * 16 + lr];
      acc = __builtin_amdgcn_wmma_f32_16x16x4_f32(false, a, false, b, (short)0, acc,
                                                  false, false);
    }
    __syncthreads();
  }
#pragma unroll
  for (int r = 0; r < 8; r++) {
    const int row = m0 + wm * 16 + r + 8 * half;   // C/D layout: lanes 16-31 -> M+8
    const int col = n0 + wn * 16 + lr;
    C[(size_t)row * NC + col] = acc[r];
  }
}

// ---------------------------------------------------------------------------
// Kernel 2: G_e = V_e^T V_e  (8 experts, 64x64 each, K=1024). One block/expert.
// Wave w computes tiles (ti0,tj) and (ti0+2,tj) -> shared B fragment.
// ---------------------------------------------------------------------------
__global__ __launch_bounds__(256) void k_gram(const float* __restrict__ V,
                                              float* __restrict__ G) {
  __shared__ float Vs[32][64];
  const int e = blockIdx.x;
  const int tid = threadIdx.x;
  const int w = tid >> 5, lane = tid & 31;
  const int half = lane >> 4, lr = lane & 15;
  const int ti0 = w >> 2, tj = w & 3, ti1 = ti0 + 2;
  const int row = tid >> 3, cb = (tid & 7) * 8;
  v8f acc0 = {}, acc1 = {};
  for (int d0 = 0; d0 < DDIM; d0 += 32) {
    const float4* src = (const float4*)&V[((size_t)e * DDIM + d0 + row) * KDIM + cb];
    float4 v0 = src[0], v1 = src[1];
    *(float4*)&Vs[row][cb]     = v0;
    *(float4*)&Vs[row][cb + 4] = v1;
    __syncthreads();
#pragma unroll
    for (int kk = 0; kk < 32; kk += 4) {
      const int krow = kk + 2 * half;
      v2f a0, a1, b;
      a0[0] = Vs[krow][ti0 * 16 + lr];  a0[1] = Vs[krow + 1][ti0 * 16 + lr];
      a1[0] = Vs[krow][ti1 * 16 + lr];  a1[1] = Vs[krow + 1][ti1 * 16 + lr];
      b[0]  = Vs[krow][tj * 16 + lr];   b[1]  = Vs[krow + 1][tj * 16 + lr];
      acc0 = __builtin_amdgcn_wmma_f32_16x16x4_f32(false, a0, false, b, (short)0, acc0,
                                                   false, false);
      acc1 = __builtin_amdgcn_wmma_f32_16x16x4_f32(false, a1, false, b, (short)0, acc1,
                                                   false, false);
    }
    __syncthreads();
  }
#pragma unroll
  for (int r = 0; r < 8; r++) {
    const int col = tj * 16 + lr;
    G[(size_t)e * (KDIM * KDIM) + (ti0 * 16 + r + 8 * half) * KDIM + col] = acc0[r];
    G[(size_t)e * (KDIM * KDIM) + (ti1 * 16 + r + 8 * half) * KDIM + col] = acc1[r];
  }
}

// ---------------------------------------------------------------------------
// Kernel 3: T[n, e*64+j] = sum_i C[n, e*64+i] * G_e[i,j]  (per-expert 8192x64x64)
// Block tile 64(M) x 64(N=full expert), K=64. Wave w -> tiles (w>>2,w&3),(+2,same).
// ---------------------------------------------------------------------------
__global__ __launch_bounds__(256) void k_gemm_cg(const float* __restrict__ C,
                                                 const float* __restrict__ G,
                                                 float* __restrict__ T) {
  __shared__ float Cs[64][68];   // pad 68: 16B-aligned rows, conflict-free frag reads
  __shared__ float Gs[64][64];
  const int m0 = blockIdx.x * 64;
  const int e  = blockIdx.y;
  const int tid = threadIdx.x;
  const int w = tid >> 5, lane = tid & 31;
  const int half = lane >> 4, lr = lane & 15;
  const int ti0 = w >> 2, tj = w & 3, ti1 = ti0 + 2;
  {
    const int row = tid >> 2, cb = (tid & 3) * 16;
    const float4* cs = (const float4*)&C[(size_t)(m0 + row) * NC + e * KDIM + cb];
    const float4* gs = (const float4*)&G[(size_t)e * (KDIM * KDIM) + row * KDIM + cb];
#pragma unroll
    for (int i = 0; i < 4; i++) {
      *(float4*)&Cs[row][cb + 4 * i] = cs[i];
      *(float4*)&Gs[row][cb + 4 * i] = gs[i];
    }
  }
  __syncthreads();
  v8f acc0 = {}, acc1 = {};
#pragma unroll
  for (int kk = 0; kk < 64; kk += 4) {
    const int krow = kk + 2 * half;
    v2f a0, a1, b;
    a0[0] = Cs[ti0 * 16 + lr][krow];  a0[1] = Cs[ti0 * 16 + lr][krow + 1];
    a1[0] = Cs[ti1 * 16 + lr][krow];  a1[1] = Cs[ti1 * 16 + lr][krow + 1];
    b[0]  = Gs[krow][tj * 16 + lr];   b[1]  = Gs[krow + 1][tj * 16 + lr];
    acc0 = __builtin_amdgcn_wmma_f32_16x16x4_f32(false, a0, false, b, (short)0, acc0,
                                                 false, false);
    acc1 = __builtin_amdgcn_wmma_f32_16x16x4_f32(false, a1, false, b, (short)0, acc1,
                                                 false, false);
  }
#pragma unroll
  for (int r = 0; r < 8; r++) {
    const int col = e * KDIM + tj * 16 + lr;
    T[(size_t)(m0 + ti0 * 16 + r + 8 * half) * NC + col] = acc0[r];
    T[(size_t)(m0 + ti1 * 16 + r + 8 * half) * NC + col] = acc1[r];
  }
}

// ---------------------------------------------------------------------------
// Kernel 4: per token (one wave each): residual_e = sqrt(|x|^2 - 2|c_e|^2 + c_e.t_e),
// softmax(-r), write routing weights R[n,8], set global "any > 1/8" flag.
// ---------------------------------------------------------------------------
__global__ __launch_bounds__(256) void k_resid(const float* __restrict__ X,
                                               const float* __restrict__ C,
                                               const float* __restrict__ T,
                                               float* __restrict__ R,
                                               unsigned int* __restrict__ flag) {
  const int w = threadIdx.x >> 5, lane = threadIdx.x & 31;
  const int n = blockIdx.x * 8 + w;
  const float* xp = &X[(size_t)n * DDIM];
  float xs = 0.f;
#pragma unroll
  for (int i = 0; i < DDIM / 32; i++) {
    float xv = xp[i * 32 + lane];
    xs += xv * xv;
  }
  for (int off = 16; off; off >>= 1) xs += __shfl_xor(xs, off, 32);

  float r[NEXP];
#pragma unroll
  for (int e = 0; e < NEXP; e++) {
    const float* cp = &C[(size_t)n * NC + e * KDIM];
    const float* tp = &T[(size_t)n * NC + e * KDIM];
    float c0 = cp[lane], c1 = cp[lane + 32];
    float t0 = tp[lane], t1 = tp[lane + 32];
    float s2 = c0 * c0 + c1 * c1;   // ||c||^2 partial
    float q  = c0 * t0 + c1 * t1;   // c G c^T partial
    for (int off = 16; off; off >>= 1) {
      s2 += __shfl_xor(s2, off, 32);
      q  += __shfl_xor(q, off, 32);
    }
    r[e] = sqrtf(fmaxf(xs - 2.f * s2 + q, 0.f));
  }
  float mn = r[0];
#pragma unroll
  for (int e = 1; e < NEXP; e++) mn = fminf(mn, r[e]);
  float wg[NEXP], sum = 0.f;
#pragma unroll
  for (int e = 0; e < NEXP; e++) { wg[e] = expf(-(r[e] - mn)); sum += wg[e]; }
  const float inv = 1.f / sum;
  bool any = false;
#pragma unroll
  for (int e = 0; e < NEXP; e++) { wg[e] *= inv; any = any || (wg[e] > 0.125f); }
  if (lane < NEXP) R[(size_t)n * NEXP + lane] = wg[lane];
  if (lane == 0 && any) atomicOr(flag, 1u);
}

// ---------------------------------------------------------------------------
// Kernel 5: threshold/top-2 mask + renormalize (needs the global flag).
// ---------------------------------------------------------------------------
__global__ void k_final(const float* __restrict__ R,
                        const unsigned int* __restrict__ flag,
                        float* __restrict__ out) {
  const int n = blockIdx.x * blockDim.x + threadIdx.x;
  if (n >= NTOK) return;
  float wv[NEXP];
#pragma unroll
  for (int e = 0; e < NEXP; e++) wv[e] = R[(size_t)n * NEXP + e];
  int i1 = 0;
#pragma unroll
  for (int e = 1; e < NEXP; e++) if (wv[e] > wv[i1]) i1 = e;     // argmax (first on tie)
  int i2 = -1;
#pragma unroll
  for (int e = 0; e < NEXP; e++) {
    if (e == i1) continue;
    if (i2 < 0 || wv[e] > wv[i2]) i2 = e;                        // 2nd best
  }
  const bool anyflag = (*flag != 0u);
  float filt[NEXP], s = 0.f;
#pragma unroll
  for (int e = 0; e < NEXP; e++) {
    bool m = anyflag ? (wv[e] > 0.125f) : (e == i1);
    m = m && (e == i1 || e == i2);                                // top-2 restriction
    filt[e] = m ? wv[e] : 0.f;
    s += filt[e];
  }
  if (s == 0.f) s = 1.f;
  const float invs = 1.f / s;
#pragma unroll
  for (int e = 0; e < NEXP; e++) out[(size_t)n * NEXP + e] = filt[e] * invs;
}

__global__ void k_init(unsigned int* flag) { *flag = 0u; }

// ---------------------------------------------------------------------------
extern "C" void kernel_launch(void* const* d_in, const int* in_sizes, int n_in,
                              void* d_out, int out_size, void* d_ws, size_t ws_size,
                              hipStream_t stream) {
  const float* x = (const float*)d_in[0];
  // d_in[1] (x_l) is unused, matching the reference.
  const float* v = (const float*)d_in[2];
  float* out = (float*)d_out;

  float* ws = (float*)d_ws;
  float* C = ws;                                   // 8192*512      = 4,194,304 f
  float* T = ws + (size_t)NTOK * NC;               // 8192*512
  float* G = ws + 2 * (size_t)NTOK * NC;           // 8*64*64       = 32,768 f
  float* R = G + (size_t)NEXP * KDIM * KDIM;       // 8192*8        = 65,536 f
  unsigned int* flag = (unsigned int*)(R + (size_t)NTOK * NEXP);

  k_init<<<1, 1, 0, stream>>>(flag);
  k_gram<<<NEXP, 256, 0, stream>>>(v, G);
  k_gemm_xv<<<dim3(NTOK / 64, NC / 32), 256, 0, stream>>>(x, v, C);
  k_gemm_cg<<<dim3(NTOK / 64, NEXP), 256, 0, stream>>>(C, G, T);
  k_resid<<<NTOK / 8, 256, 0, stream>>>(x, C, T, R, flag);
  k_final<<<(NTOK + 255) / 256, 256, 0, stream>>>(R, flag, out);
}